// Train_GIoU_3667902070874
// MI455X (gfx1250) — compile-verified
//
#include <hip/hip_runtime.h>
#include <cstdint>
#include <cfloat>
#include <cmath>

// Problem constants (match reference)
#define Hh        256
#define Ww        256
#define HW        65536          // H*W
#define Bb        4
#define NT        9
#define Ss        (HW + 1)       // segment count S = H*W+1
#define NB        (Bb * NT)      // 36 blocks for main kernel
#define NTHREADS  1024

// LDS layout (byte offsets into dynamic shared memory)
#define LAB_OFF    0             // uint32 lab[HW]            = 262144 B
#define REDF_OFF   262144        // float  redF[1024]         =   4096 B
#define REDU_OFF   266240        // uint32/int redU[1024]     =   4096 B
#define REDG_OFF   270336        // float  redG[1024]         =   4096 B
#define SCAL_OFF   274432        // scalars (chg flag)        =     64 B
#define STAGE_OFF  274496        // 2 x 16384 async staging   =  32768 B
#define CHB        16384         // bytes per chunk (4096 floats)
#define NCH        16            // 16 chunks cover 65536 floats
#define SMEM_BYTES (STAGE_OFF + 2 * CHB)   // 307264 B <= 320 KB LDS/WGP

static_assert(SMEM_BYTES <= 320 * 1024, "exceeds CDNA5 LDS per workgroup");

__device__ __constant__ float c_thr[NT] = {0.9f, 0.8f, 0.7f, 0.6f, 0.5f,
                                           0.4f, 0.3f, 0.2f, 0.1f};

__device__ __forceinline__ int   imin(int a, int b)   { return a < b ? a : b; }
__device__ __forceinline__ int   imax(int a, int b)   { return a > b ? a : b; }

// ---- block-wide reductions over 1024 threads (scratch = 1024 entries) ----
__device__ __forceinline__ float blockMinF(float v, float* red, int tid) {
  red[tid] = v; __syncthreads();
  for (int o = NTHREADS / 2; o > 0; o >>= 1) {
    if (tid < o) red[tid] = fminf(red[tid], red[tid + o]);
    __syncthreads();
  }
  float r = red[0]; __syncthreads(); return r;
}
__device__ __forceinline__ float blockMaxF(float v, float* red, int tid) {
  red[tid] = v; __syncthreads();
  for (int o = NTHREADS / 2; o > 0; o >>= 1) {
    if (tid < o) red[tid] = fmaxf(red[tid], red[tid + o]);
    __syncthreads();
  }
  float r = red[0]; __syncthreads(); return r;
}
__device__ __forceinline__ float blockSumF(float v, float* red, int tid) {
  red[tid] = v; __syncthreads();
  for (int o = NTHREADS / 2; o > 0; o >>= 1) {
    if (tid < o) red[tid] = red[tid] + red[tid + o];
    __syncthreads();
  }
  float r = red[0]; __syncthreads(); return r;
}
__device__ __forceinline__ int blockMinI(int v, int* red, int tid) {
  red[tid] = v; __syncthreads();
  for (int o = NTHREADS / 2; o > 0; o >>= 1) {
    if (tid < o) red[tid] = imin(red[tid], red[tid + o]);
    __syncthreads();
  }
  int r = red[0]; __syncthreads(); return r;
}
__device__ __forceinline__ int blockMaxI(int v, int* red, int tid) {
  red[tid] = v; __syncthreads();
  for (int o = NTHREADS / 2; o > 0; o >>= 1) {
    if (tid < o) red[tid] = imax(red[tid], red[tid + o]);
    __syncthreads();
  }
  int r = red[0]; __syncthreads(); return r;
}

// CDNA5 async global->LDS copy: 16 bytes/lane, GVS addressing (saddr + voffset).
__device__ __forceinline__ void asyncCopy16(uint32_t ldsByteAddr,
                                            const void* sbase,
                                            uint32_t vByteOff) {
  asm volatile("global_load_async_to_lds_b128 %0, %1, %2"
               :
               : "v"(ldsByteAddr), "v"(vByteOff), "s"(sbase)
               : "memory");
}

// One block per (image b, threshold t).  Everything CCL-related lives in LDS.
__global__ void __launch_bounds__(NTHREADS)
giou_ccl_kernel(const float* __restrict__ fake, const float* __restrict__ real,
                uint32_t* __restrict__ ws) {
  extern __shared__ char smem[];
  uint32_t* lab  = (uint32_t*)(smem + LAB_OFF);
  float*    redF = (float*)(smem + REDF_OFF);
  int*      redI = (int*)(smem + REDU_OFF);
  uint32_t* redU = (uint32_t*)(smem + REDU_OFF);
  float*    redG = (float*)(smem + REDG_OFF);
  volatile uint32_t* chg = (volatile uint32_t*)(smem + SCAL_OFF);

  const int tid = threadIdx.x;
  const int bt  = blockIdx.x;
  const int b   = bt / NT;
  const int t   = bt % NT;

  const float* gf = fake + (size_t)b * HW;
  const float* gr = real + (size_t)b * HW;

  // ---------- Phase A1: async-stream fake image through LDS; min/max ----------
  float mn = FLT_MAX, mx = -FLT_MAX;
  {
    const uint32_t ldsBase = (uint32_t)STAGE_OFF + (uint32_t)tid * 16u;
    const uint32_t gBase   = (uint32_t)tid * 16u;
    asyncCopy16(ldsBase, gf, gBase);                       // chunk 0 -> buf 0
    for (int c = 0; c < NCH; ++c) {
      if (c + 1 < NCH) {
        asyncCopy16(ldsBase + (uint32_t)(((c + 1) & 1) * CHB),
                    gf, gBase + (uint32_t)((c + 1) * CHB)); // prefetch next
        asm volatile("s_wait_asynccnt 1" ::: "memory");     // chunk c complete
      } else {
        asm volatile("s_wait_asynccnt 0" ::: "memory");
      }
      __syncthreads();
      const float4 v =
          *(const float4*)(smem + STAGE_OFF + (c & 1) * CHB + tid * 16);
      mn = fminf(mn, fminf(fminf(v.x, v.y), fminf(v.z, v.w)));
      mx = fmaxf(mx, fmaxf(fmaxf(v.x, v.y), fmaxf(v.z, v.w)));
      __syncthreads();   // all reads done before this buffer is reused
    }
  }
  const float fmn = blockMinF(mn, redF, tid);
  const float fmx = blockMaxF(mx, redF, tid);

  // ---------- Phase A2: GT bbox + T_real from real image ----------
  int rmn = Hh, rmx = -1, cmn = Ww, cmx = -1;
  float tsum = 0.f;
  for (int i = tid; i < HW; i += NTHREADS) {
    if (i + NTHREADS < HW) __builtin_prefetch(gr + i + NTHREADS, 0, 0);
    float v = gr[i];
    if (v != 0.0f) {
      int r = i >> 8, c = i & 255;
      rmn = imin(rmn, r); rmx = imax(rmx, r);
      cmn = imin(cmn, c); cmx = imax(cmx, c);
      tsum += v;
    }
  }
  const float treal = blockSumF(tsum, redF, tid);
  const float g0 = (float)blockMinI(rmn, redI, tid);   // gt row-min
  const float g2 = (float)blockMaxI(rmx, redI, tid);   // gt row-max
  const float g1 = (float)blockMinI(cmn, redI, tid);   // gt col-min
  const float g3 = (float)blockMaxI(cmx, redI, tid);   // gt col-max
  const float wG = g2 - g0, hG = g3 - g1;
  const float areag =
      ((wG == 0.f) || (hG == 0.f)) ? (wG + 1.f) * (hG + 1.f) : wG * hG;

  // ---------- Label init for this threshold ----------
  // f>(t) <=> fake > mn + t*(mx-mn)
  const float cut = fmn + c_thr[t] * (fmx - fmn);
  for (int i = tid; i < HW; i += NTHREADS) {
    lab[i] = (gf[i] > cut) ? (uint32_t)(i + 1) : 0u;
  }
  __syncthreads();

  // ---------- CCL: in-place 8-neighbor min flood + pointer jumping ----------
  for (int iter = 0; iter < 8192; ++iter) {
    if (tid == 0) *chg = 0u;
    __syncthreads();
    uint32_t my = 0u;
    for (int i = tid; i < HW; i += NTHREADS) {
      uint32_t l = lab[i];
      if (!l) continue;
      const int r = i >> 8, c = i & 255;
      uint32_t m = l;
      if (r > 0) {
        const int j = i - 256;
        if (c > 0)   { uint32_t x = lab[j - 1]; if (x && x < m) m = x; }
        { uint32_t x = lab[j];     if (x && x < m) m = x; }
        if (c < 255) { uint32_t x = lab[j + 1]; if (x && x < m) m = x; }
      }
      if (c > 0)   { uint32_t x = lab[i - 1]; if (x && x < m) m = x; }
      if (c < 255) { uint32_t x = lab[i + 1]; if (x && x < m) m = x; }
      if (r < 255) {
        const int j = i + 256;
        if (c > 0)   { uint32_t x = lab[j - 1]; if (x && x < m) m = x; }
        { uint32_t x = lab[j];     if (x && x < m) m = x; }
        if (c < 255) { uint32_t x = lab[j + 1]; if (x && x < m) m = x; }
      }
      // pointer jump toward component root (labels only ever decrease)
      uint32_t q = lab[m - 1];
      int hop = 0;
      while (q && q < m && hop < 64) { m = q; q = lab[m - 1]; ++hop; }
      if (m < l) { lab[i] = m; my = 1u; }
    }
    if (my) *chg = 1u;
    __syncthreads();
    const uint32_t done = (*chg == 0u);
    __syncthreads();
    if (done) break;
  }

  // ---------- Per-component stats into this block's global scratch ----------
  uint32_t* myws  = ws + (size_t)bt * (size_t)(6 * (size_t)Ss);
  uint32_t* s_cnt = myws;
  uint32_t* s_itr = myws + (size_t)Ss;
  uint32_t* s_rmn = myws + 2 * (size_t)Ss;
  uint32_t* s_rmx = myws + 3 * (size_t)Ss;
  uint32_t* s_cmn = myws + 4 * (size_t)Ss;
  uint32_t* s_cmx = myws + 5 * (size_t)Ss;
  for (int s = tid; s < Ss; s += NTHREADS) {
    s_cnt[s] = 0u; s_itr[s] = 0u;
    s_rmn[s] = 0xFFFFFFFFu; s_rmx[s] = 0u;
    s_cmn[s] = 0xFFFFFFFFu; s_cmx[s] = 0u;
  }
  __threadfence();
  __syncthreads();
  for (int i = tid; i < HW; i += NTHREADS) {
    uint32_t l = lab[i];
    if (!l) continue;
    const uint32_t r = (uint32_t)i >> 8, c = (uint32_t)i & 255u;
    atomicAdd(&s_cnt[l], 1u);
    if (gr[i] != 0.0f) atomicAdd(&s_itr[l], 1u);
    atomicMin(&s_rmn[l], r); atomicMax(&s_rmx[l], r);
    atomicMin(&s_cmn[l], c); atomicMax(&s_cmx[l], c);
  }
  __threadfence();
  __syncthreads();

  // ---------- Evaluate components: argmax IoU (smallest label tiebreak) ----
  float bIou = -INFINITY, bGiou = 0.f, bDice = -INFINITY;
  uint32_t bS = 0xFFFFFFFFu;
  for (int s = tid; s < Ss; s += NTHREADS) {
    if (s == 0) continue;
    const uint32_t cn = s_cnt[s];
    if (!cn) continue;
    const float cntf  = (float)cn;
    const float inr   = (float)s_itr[s];
    const float rmnf  = (float)(int)s_rmn[s];
    const float rmxf  = (float)(int)s_rmx[s];
    const float cmnf  = (float)(int)s_cmn[s];
    const float cmxf  = (float)(int)s_cmx[s];
    const float dice  = (2.f * inr + 1.f) / (cntf + treal + 1.f);
    const float w = rmxf - rmnf, h = cmxf - cmnf;
    const float areap =
        ((w == 0.f) || (h == 0.f)) ? (w + 1.f) * (h + 1.f) : w * h;
    const float xI = fmaxf(fminf(rmxf, g2) - fmaxf(rmnf, g0), 0.f);
    const float yI = fmaxf(fminf(cmxf, g3) - fmaxf(cmnf, g1), 0.f);
    const float inter = xI * yI;
    const float carea = (fmaxf(rmxf, g2) - fminf(rmnf, g0)) *
                        (fmaxf(cmxf, g3) - fminf(cmnf, g1));
    const float uni  = areap + areag - inter;
    const float iou  = inter / uni;
    const float giou = iou - (carea - uni) / carea;
    if (dice > bDice) bDice = dice;
    if (iou > bIou || (iou == bIou && (uint32_t)s < bS)) {
      bIou = iou; bS = (uint32_t)s; bGiou = giou;
    }
  }

  // reduce (iou, label, giou) with argmax + smallest-label tie-break
  redF[tid] = bIou; redU[tid] = bS; redG[tid] = bGiou;
  __syncthreads();
  for (int o = NTHREADS / 2; o > 0; o >>= 1) {
    if (tid < o) {
      const float f2 = redF[tid + o];
      if (f2 > redF[tid] || (f2 == redF[tid] && redU[tid + o] < redU[tid])) {
        redF[tid] = f2; redU[tid] = redU[tid + o]; redG[tid] = redG[tid + o];
      }
    }
    __syncthreads();
  }
  const float wIou = redF[0], wGiou = redG[0];
  __syncthreads();
  redF[tid] = bDice; __syncthreads();
  for (int o = NTHREADS / 2; o > 0; o >>= 1) {
    if (tid < o) redF[tid] = fmaxf(redF[tid], redF[tid + o]);
    __syncthreads();
  }
  if (tid == 0) {
    float* res = (float*)(ws + (size_t)NB * (size_t)(6 * (size_t)Ss));
    res[bt * 4 + 0] = wIou;      // -inf => no valid component
    res[bt * 4 + 1] = 0.f;
    res[bt * 4 + 2] = wGiou;
    res[bt * 4 + 3] = redF[0];   // dice max (-inf if none)
  }
}

// Combine the 9 per-threshold results per image; thresholds in ascending index
// order with strict '>' reproduces jnp.argmax first-occurrence tie-breaking.
__global__ void giou_finalize_kernel(const uint32_t* __restrict__ ws,
                                     float* __restrict__ out) {
  const int b = threadIdx.x;
  if (b >= Bb) return;
  const float* res = (const float*)(ws + (size_t)NB * (size_t)(6 * (size_t)Ss));
  float bi = -INFINITY, bg = 0.f, bthr = 0.f, bd = -INFINITY;
  bool any = false;
  for (int t = 0; t < NT; ++t) {
    const float* r = res + (size_t)(b * NT + t) * 4;
    const float iou = r[0], giou = r[2], dice = r[3];
    if (!(iou > -INFINITY)) continue;   // invalid threshold (no components)
    any = true;
    if (iou > bi) { bi = iou; bg = giou; bthr = c_thr[t]; }
    if (dice > bd) bd = dice;
  }
  if (!any) { bg = 0.f; bthr = 0.f; bd = 0.f; }
  out[0 * Bb + b] = 1.f - bg;   // loss_giou
  out[1 * Bb + b] = bg;         // giou
  out[2 * Bb + b] = bthr;       // threshold
  out[3 * Bb + b] = bd;         // dice
}

extern "C" void kernel_launch(void* const* d_in, const int* in_sizes, int n_in,
                              void* d_out, int out_size, void* d_ws,
                              size_t ws_size, hipStream_t stream) {
  (void)in_sizes; (void)n_in; (void)out_size; (void)ws_size;
  const float* fake = (const float*)d_in[0];
  const float* real = (const float*)d_in[1];
  float* out   = (float*)d_out;
  uint32_t* ws = (uint32_t*)d_ws;   // needs 36 * 65537 * 6 * 4 + 576 B ~= 56.6 MB

  (void)hipFuncSetAttribute((const void*)giou_ccl_kernel,
                            hipFuncAttributeMaxDynamicSharedMemorySize,
                            SMEM_BYTES);
  giou_ccl_kernel<<<NB, NTHREADS, SMEM_BYTES, stream>>>(fake, real, ws);
  giou_finalize_kernel<<<1, 32, 0, stream>>>(ws, out);
}